// CorotationalBeam2D_12146167513813
// MI455X (gfx1250) — compile-verified
//
#include <hip/hip_runtime.h>
#include <hip/hip_bf16.h>

typedef float v2f __attribute__((ext_vector_type(2)));
typedef int   v2i __attribute__((ext_vector_type(2)));

struct __attribute__((packed)) F3 { float x, y, z; };  // 12B row, align 4

// ---------------------------------------------------------------------------
// Zero the nodal_forces accumulation region (d_out is poisoned by harness).
// ---------------------------------------------------------------------------
__global__ __launch_bounds__(256) void zero_nf_kernel(float* __restrict__ p, int n) {
    int i = blockIdx.x * blockDim.x + threadIdx.x;
    if (i < n) p[i] = 0.0f;
}

// ---------------------------------------------------------------------------
// One thread per beam element: async-stage conn through LDS (CDNA5 ASYNCcnt
// path), gather endpoint rows as b96, corotational transform, local force
// evaluation, rotate back, scatter-add nodal forces, stream all per-element
// fields out with non-temporal stores.
// ---------------------------------------------------------------------------
__global__ __launch_bounds__(256) void beam_kernel(
    const float* __restrict__ pred_disp,   // (n_nodes, 3)
    const float* __restrict__ coords,      // (n_nodes, 3)
    const int*   __restrict__ conn,        // (n_elems, 2)
    const float* __restrict__ prop_E,      // (n_elems,)
    const float* __restrict__ prop_A,      // (n_elems,)
    const float* __restrict__ prop_I22,    // (n_elems,)
    float* __restrict__ out,
    int n_elems, int n_nodes)
{
    __shared__ int s_conn[512];            // 256 elements x {nA,nB}

    const int tid = threadIdx.x;
    const int e   = blockIdx.x * blockDim.x + tid;
    if (e >= n_elems) return;

    // ---- output region offsets (flat concatenation, reference return order)
    const size_t NE   = (size_t)n_elems;
    const size_t o_nf = 0;                      // nodal_forces  (3*n_nodes)
    const size_t o_fg = (size_t)3 * n_nodes;    // f_global      (6*NE)
    const size_t o_fl = o_fg + 6 * NE;          // f_local       (6*NE)
    const size_t o_dl = o_fl + 6 * NE;          // d_local       (6*NE)
    const size_t o_dg = o_dl + 6 * NE;          // d_global      (6*NE)
    const size_t o_N  = o_dg + 6 * NE;          // N_e
    const size_t o_V  = o_N  + NE;              // V_e
    const size_t o_M1 = o_V  + NE;              // M1_e
    const size_t o_M2 = o_M1 + NE;              // M2_e
    const size_t o_ul = o_M2 + NE;              // u_l
    const size_t o_ta = o_ul + NE;              // ta
    const size_t o_tb = o_ta + NE;              // tb
    const size_t o_ph = o_tb + NE;              // phi
    const size_t o_l0 = o_ph + NE;              // l0
    const size_t o_c  = o_l0 + NE;              // c
    const size_t o_s  = o_c  + NE;              // s

    // ---- CDNA5 async copy: each lane drops its own 8B conn pair into LDS
    //      (GVS mode: SGPR base + 32-bit VGPR byte offset). Tracked by
    //      ASYNCcnt; each thread reads back only its own slot, so a
    //      wave-level s_wait_asynccnt is sufficient (no barrier needed).
    {
        unsigned lds_addr = (unsigned)(uintptr_t)&s_conn[2 * tid];
        unsigned goff     = (unsigned)((size_t)e * 8u);
        asm volatile("global_load_async_to_lds_b64 %0, %1, %2"
                     :
                     : "v"(lds_addr), "v"(goff), "s"(conn)
                     : "memory");
    }

    // ---- streaming (read-once) property inputs: non-temporal loads keep
    //      them out of L2 so the 96MB coords+pred_disp gather set stays
    //      resident.
    const float E  = __builtin_nontemporal_load(prop_E   + e);
    const float A  = __builtin_nontemporal_load(prop_A   + e);
    const float I2 = __builtin_nontemporal_load(prop_I22 + e);

    asm volatile("s_wait_asynccnt 0x0" ::: "memory");
    const v2i cn = *(const v2i*)&s_conn[2 * tid];   // ds_load_b64
    const int nA = cn.x;
    const int nB = cn.y;

    // CDNA5 speculative prefetch of the gathered rows (global_prefetch_b8)
    __builtin_prefetch(pred_disp + 3 * (size_t)nA, 0, 3);
    __builtin_prefetch(pred_disp + 3 * (size_t)nB, 0, 3);

    // ---- gathered (L2-resident) inputs: full 12B rows -> global_load_b96
    const F3 cA = *(const F3*)(coords    + 3 * (size_t)nA);
    const F3 cB = *(const F3*)(coords    + 3 * (size_t)nB);
    const F3 dA = *(const F3*)(pred_disp + 3 * (size_t)nA);
    const F3 dB = *(const F3*)(pred_disp + 3 * (size_t)nB);

    // ---- geometry
    const float dx0 = cB.x - cA.x;
    const float dz0 = cB.z - cA.z;
    const float l0  = sqrtf(dx0 * dx0 + dz0 * dz0);
    const float iL  = 1.0f / l0;          // one precise IEEE divide, reused
    const float c   = dx0 * iL;
    const float s   = dz0 * iL;

    // ---- local displacements
    const float ua =  c * dA.x + s * dA.y;
    const float wa = -s * dA.x + c * dA.y;
    const float ta =  dA.z;
    const float ub =  c * dB.x + s * dB.y;
    const float wb = -s * dB.x + c * dB.y;
    const float tb =  dB.z;

    // ---- local forces
    const float EA  = E * A;
    const float EI  = E * I2;
    const float iL2 = iL * iL;
    const float iL3 = iL2 * iL;
    const float EA_L    = EA * iL;
    const float EI_L    = EI * iL;
    const float EI6_L2  = 6.0f  * EI * iL2;
    const float EI12_L3 = 12.0f * EI * iL3;

    const float wab = wa - wb;
    const float tab = ta + tb;

    const float f0 = EA_L * (ua - ub);
    const float f1 = EI12_L3 * wab + EI6_L2 * tab;
    const float f2 = EI6_L2 * wab + EI_L * (4.0f * ta + 2.0f * tb);
    const float f3 = EA_L * (ub - ua);
    const float f4 = EI12_L3 * (wb - wa) - EI6_L2 * tab;
    const float f5 = EI6_L2 * wab + EI_L * (2.0f * ta + 4.0f * tb);

    // ---- rotate back to global frame
    const float g0 = c * f0 - s * f1;
    const float g1 = s * f0 + c * f1;
    const float g3 = c * f3 - s * f4;
    const float g4 = s * f3 + c * f4;

    // ---- scatter-add nodal forces (hardware global_atomic_add_f32, L2-resident)
    {
        float* nf = out + o_nf;
        unsafeAtomicAdd(nf + 3 * (size_t)nA + 0, g0);
        unsafeAtomicAdd(nf + 3 * (size_t)nA + 1, g1);
        unsafeAtomicAdd(nf + 3 * (size_t)nA + 2, f2);
        unsafeAtomicAdd(nf + 3 * (size_t)nB + 0, g3);
        unsafeAtomicAdd(nf + 3 * (size_t)nB + 1, g4);
        unsafeAtomicAdd(nf + 3 * (size_t)nB + 2, f5);
    }

    // ---- streaming outputs: non-temporal 8B vector stores (rows are 24B,
    //      8B-aligned), protecting L2 from 568MB of write-once traffic.
    {
        v2f* fg = (v2f*)(out + o_fg) + 3 * (size_t)e;
        __builtin_nontemporal_store((v2f){g0, g1}, fg + 0);
        __builtin_nontemporal_store((v2f){f2, g3}, fg + 1);
        __builtin_nontemporal_store((v2f){g4, f5}, fg + 2);

        v2f* fl = (v2f*)(out + o_fl) + 3 * (size_t)e;
        __builtin_nontemporal_store((v2f){f0, f1}, fl + 0);
        __builtin_nontemporal_store((v2f){f2, f3}, fl + 1);
        __builtin_nontemporal_store((v2f){f4, f5}, fl + 2);

        v2f* dl = (v2f*)(out + o_dl) + 3 * (size_t)e;
        __builtin_nontemporal_store((v2f){ua, wa}, dl + 0);
        __builtin_nontemporal_store((v2f){ta, ub}, dl + 1);
        __builtin_nontemporal_store((v2f){wb, tb}, dl + 2);

        v2f* dg = (v2f*)(out + o_dg) + 3 * (size_t)e;
        __builtin_nontemporal_store((v2f){dA.x, dA.y}, dg + 0);
        __builtin_nontemporal_store((v2f){dA.z, dB.x}, dg + 1);
        __builtin_nontemporal_store((v2f){dB.y, dB.z}, dg + 2);
    }

    __builtin_nontemporal_store(f3,              out + o_N  + e);
    __builtin_nontemporal_store(f4,              out + o_V  + e);
    __builtin_nontemporal_store(f2,              out + o_M1 + e);
    __builtin_nontemporal_store(f5,              out + o_M2 + e);
    __builtin_nontemporal_store(ub - ua,         out + o_ul + e);
    __builtin_nontemporal_store(ta,              out + o_ta + e);
    __builtin_nontemporal_store(tb,              out + o_tb + e);
    __builtin_nontemporal_store((wb - wa) * iL,  out + o_ph + e);
    __builtin_nontemporal_store(l0,              out + o_l0 + e);
    __builtin_nontemporal_store(c,               out + o_c  + e);
    __builtin_nontemporal_store(s,               out + o_s  + e);
}

// ---------------------------------------------------------------------------
extern "C" void kernel_launch(void* const* d_in, const int* in_sizes, int n_in,
                              void* d_out, int out_size, void* d_ws, size_t ws_size,
                              hipStream_t stream) {
    const float* pred_disp = (const float*)d_in[0];
    const float* coords    = (const float*)d_in[1];
    const int*   conn      = (const int*)  d_in[2];
    const float* prop_E    = (const float*)d_in[3];
    const float* prop_A    = (const float*)d_in[4];
    const float* prop_I22  = (const float*)d_in[5];

    const int n_nodes = in_sizes[0] / 3;   // pred_disp is (n_nodes, 3)
    const int n_elems = in_sizes[3];       // prop_E is (n_elems,)

    float* out = (float*)d_out;

    // 1) zero the nodal_forces accumulator region
    const int nf = 3 * n_nodes;
    zero_nf_kernel<<<(nf + 255) / 256, 256, 0, stream>>>(out, nf);

    // 2) per-element evaluation + scatter
    beam_kernel<<<(n_elems + 255) / 256, 256, 0, stream>>>(
        pred_disp, coords, conn, prop_E, prop_A, prop_I22,
        out, n_elems, n_nodes);
}